// GlobalModel_24773371363900
// MI455X (gfx1250) — compile-verified
//
#include <hip/hip_runtime.h>

// ext-vector types (native clang vectors so __builtin_nontemporal_load works)
typedef float v2f __attribute__((ext_vector_type(2)));
typedef float v4f __attribute__((ext_vector_type(4)));
typedef float v8f __attribute__((ext_vector_type(8)));

#define NUM_GRAPHS 1024
#define DNODE 128  // node feature dim == global dim == hidden dim

// ---------------------------------------------------------------------------
// Kernel 0: zero the workspace accumulators (seg_sum [1024x128] + counts[1024])
// ---------------------------------------------------------------------------
__global__ void zero_ws_kernel(float* __restrict__ ws, int n) {
    int i = blockIdx.x * blockDim.x + threadIdx.x;
    if (i < n) ws[i] = 0.0f;
}

// ---------------------------------------------------------------------------
// Kernel 1: segment sum + counts over sorted batch ids (the roofline kernel:
// 1.02 GB of x streamed once -> ~44us floor at 23.3 TB/s).
//
// One wave owns a contiguous 256-row chunk; lane l holds cols [4l,4l+4) in
// register accumulators. Because `batch` is sorted, ~87% of 256-row chunks
// fall entirely inside one graph (avg segment ~1953 rows):
//   fast path : whole chunk is one segment -> branch-free 4x-unrolled stream
//               with 4 independent accumulators (deep clause of NT b128
//               loads in flight, no per-row id load, one atomic flush).
//   slow path : chunk straddles a boundary -> per-row run-length scan,
//               flush at each id change.
// ---------------------------------------------------------------------------
__device__ __forceinline__ void flush_seg(float* __restrict__ seg,
                                          float* __restrict__ cnt,
                                          int g, int lane, v4f acc, int run) {
    float* p = seg + (long long)g * DNODE + lane * 4;
    atomicAdd(p + 0, acc.x);
    atomicAdd(p + 1, acc.y);
    atomicAdd(p + 2, acc.z);
    atomicAdd(p + 3, acc.w);
    if (lane == 0) atomicAdd(cnt + g, (float)run);
}

__global__ __launch_bounds__(256) void segsum_kernel(
    const float* __restrict__ x, const int* __restrict__ batch,
    float* __restrict__ seg, float* __restrict__ cnt,
    int N, int rows_per_wave) {
    const int wave = (blockIdx.x * blockDim.x + threadIdx.x) >> 5;
    const int lane = threadIdx.x & 31;

    long long row0 = (long long)wave * rows_per_wave;
    if (row0 >= N) return;
    long long rowEnd = row0 + rows_per_wave;
    if (rowEnd > N) rowEnd = N;

    const int gFirst = batch[row0];
    const int gLast  = batch[rowEnd - 1];

    if (gFirst == gLast) {
        // ---- fast path: single segment, pure streaming ----
        v4f a0 = {0.f, 0.f, 0.f, 0.f};
        v4f a1 = {0.f, 0.f, 0.f, 0.f};
        v4f a2 = {0.f, 0.f, 0.f, 0.f};
        v4f a3 = {0.f, 0.f, 0.f, 0.f};
        const v4f* px = (const v4f*)(x + row0 * (long long)DNODE) + lane;
        long long r = row0;
        for (; r + 4 <= rowEnd; r += 4) {
            // 4 independent NT b128 loads -> deep memory pipeline
            v4f v0 = __builtin_nontemporal_load(px + 0 * 32);
            v4f v1 = __builtin_nontemporal_load(px + 1 * 32);
            v4f v2 = __builtin_nontemporal_load(px + 2 * 32);
            v4f v3 = __builtin_nontemporal_load(px + 3 * 32);
            a0 += v0; a1 += v1; a2 += v2; a3 += v3;
            px += 4 * 32;  // 4 rows of 128 floats
        }
        for (; r < rowEnd; ++r) {
            a0 += __builtin_nontemporal_load(px);
            px += 32;
        }
        v4f acc = (a0 + a1) + (a2 + a3);
        flush_seg(seg, cnt, gFirst, lane, acc, (int)(rowEnd - row0));
    } else {
        // ---- slow path: chunk straddles >=1 segment boundary ----
        v4f acc = {0.f, 0.f, 0.f, 0.f};
        int cur = -1;
        int run = 0;
        for (long long r = row0; r < rowEnd; ++r) {
            int g = batch[r];  // uniform across wave; L0-resident line
            if (g != cur) {
                if (cur >= 0) flush_seg(seg, cnt, cur, lane, acc, run);
                acc = (v4f){0.f, 0.f, 0.f, 0.f};
                run = 0;
                cur = g;
            }
            const v4f* px = (const v4f*)(x + r * (long long)DNODE) + lane;
            acc += __builtin_nontemporal_load(px);
            ++run;
        }
        if (cur >= 0) flush_seg(seg, cnt, cur, lane, acc, run);
    }
}

// ---------------------------------------------------------------------------
// Kernel 2: fused MLP with f32 WMMA (V_WMMA_F32_16X16X4_F32).
//   Hin = [u | seg/max(cnt,1)]            [16 rows x 256] per workgroup (LDS)
//   H   = relu(Hin @ W1 + b1)             [16 x 128]      (LDS)
//   out = H @ W2 + b2                     [16 x 128]      (global)
// 8 waves per workgroup; wave w owns output columns [16w, 16w+16).
// A-frag (16x4 f32, 2 VGPR): m = lane&15, K-half = 2*(lane>=16).
// C/D v8f: row = i + 8*(lane>=16), col = lane&15.
// GEMMs total ~100 MFLOP (negligible vs the 1 GB stream) -> keep full f32
// precision via the 16x16x4 f32 WMMA rather than downconverting.
// ---------------------------------------------------------------------------
__global__ __launch_bounds__(256) void mlp_wmma_kernel(
    const float* __restrict__ u, const float* __restrict__ seg,
    const float* __restrict__ cnt,
    const float* __restrict__ W1, const float* __restrict__ b1,
    const float* __restrict__ W2, const float* __restrict__ b2,
    float* __restrict__ out) {
    __shared__ float sHin[16][260];  // pad 256->260: M-stride hits 16 banks
    __shared__ float sH[16][132];    // pad 128->132

    const int tile = blockIdx.x;   // 16-row tile of the 1024 graphs
    const int tid  = threadIdx.x;

    // Phase 0: build [u | mean] rows into LDS. 4096 elems / 256 threads.
    for (int e = tid; e < 16 * 256; e += 256) {
        const int c = e & 255;       // = tid (coalesced along columns)
        const int r = e >> 8;
        const int g = tile * 16 + r;
        float v;
        if (c < DNODE) {
            v = u[g * DNODE + c];
        } else {
            float cc = cnt[g];
            v = seg[g * DNODE + (c - DNODE)] / fmaxf(cc, 1.0f);
        }
        sHin[r][c] = v;
    }
    __syncthreads();

    const int w     = tid >> 5;     // wave 0..7 -> N-slab
    const int lane  = tid & 31;
    const int nl    = lane & 15;    // col-in-slab / A-row
    const int hi    = lane >> 4;    // 0 | 1
    const int khalf = hi * 2;       // K offset for A/B frags
    const int n     = w * 16 + nl;  // global column 0..127

    // ---- GEMM1: H = relu(Hin @ W1 + b1), K = 256 ----
    v8f acc;
#pragma unroll
    for (int i = 0; i < 8; ++i) acc[i] = b1[n];

#pragma unroll 4
    for (int kb = 0; kb < 256; kb += 4) {
        v2f a, b;
        a[0] = sHin[nl][kb + khalf];
        a[1] = sHin[nl][kb + khalf + 1];
        b[0] = W1[(kb + khalf) * DNODE + n];
        b[1] = W1[(kb + khalf + 1) * DNODE + n];
        acc = __builtin_amdgcn_wmma_f32_16x16x4_f32(
            /*neg_a=*/false, a, /*neg_b=*/false, b,
            /*c_mod=*/(short)0, acc, /*reuse_a=*/false, /*reuse_b=*/false);
    }

#pragma unroll
    for (int i = 0; i < 8; ++i) {
        sH[i + hi * 8][n] = fmaxf(acc[i], 0.0f);
    }
    __syncthreads();

    // ---- GEMM2: out = H @ W2 + b2, K = 128 ----
    v8f acc2;
#pragma unroll
    for (int i = 0; i < 8; ++i) acc2[i] = b2[n];

#pragma unroll 4
    for (int kb = 0; kb < 128; kb += 4) {
        v2f a, b;
        a[0] = sH[nl][kb + khalf];
        a[1] = sH[nl][kb + khalf + 1];
        b[0] = W2[(kb + khalf) * DNODE + n];
        b[1] = W2[(kb + khalf + 1) * DNODE + n];
        acc2 = __builtin_amdgcn_wmma_f32_16x16x4_f32(
            false, a, false, b, (short)0, acc2, false, false);
    }

#pragma unroll
    for (int i = 0; i < 8; ++i) {
        const int row = tile * 16 + i + hi * 8;
        out[row * DNODE + n] = acc2[i];
    }
}

// ---------------------------------------------------------------------------
// Host-side launcher.
// Inputs (setup_inputs order): 0:x 1:edge_index 2:edge_attr 3:u 4:batch
//                              5:W1 6:b1 7:W2 8:b2   (1,2 unused by reference)
// ---------------------------------------------------------------------------
extern "C" void kernel_launch(void* const* d_in, const int* in_sizes, int n_in,
                              void* d_out, int out_size, void* d_ws, size_t ws_size,
                              hipStream_t stream) {
    const float* x     = (const float*)d_in[0];
    const float* u     = (const float*)d_in[3];
    const int*   batch = (const int*)d_in[4];
    const float* W1    = (const float*)d_in[5];
    const float* b1    = (const float*)d_in[6];
    const float* W2    = (const float*)d_in[7];
    const float* b2    = (const float*)d_in[8];
    float*       out   = (float*)d_out;

    const int N = in_sizes[0] / DNODE;  // 2,000,000 nodes

    float* seg = (float*)d_ws;                   // [1024 x 128]
    float* cnt = seg + NUM_GRAPHS * DNODE;       // [1024]

    // 1) zero accumulators (harness poisons ws -> must re-init each call)
    const int ztot = NUM_GRAPHS * DNODE + NUM_GRAPHS;
    zero_ws_kernel<<<(ztot + 255) / 256, 256, 0, stream>>>(seg, ztot);

    // 2) bandwidth-bound segment sum
    const int rows_per_wave = 256;
    const long long waves = ((long long)N + rows_per_wave - 1) / rows_per_wave;
    const int blocks = (int)((waves + 7) / 8);  // 8 waves per 256-thread block
    segsum_kernel<<<blocks, 256, 0, stream>>>(x, batch, seg, cnt, N, rows_per_wave);

    // 3) WMMA MLP: 64 workgroups x 16 rows
    mlp_wmma_kernel<<<NUM_GRAPHS / 16, 256, 0, stream>>>(
        u, seg, cnt, W1, b1, W2, b2, out);
}